// GATLayer_25314537242829
// MI455X (gfx1250) — compile-verified
//
#include <hip/hip_runtime.h>
#include <hip/hip_bf16.h>
#include <math.h>

// GAT layer, MI455X (gfx1250, wave32).
// Bandwidth-bound: adj read (134MB) + attention write (134MB) ~= 11.5us at
// 23.3 TB/s. Everything stays fp32; both GEMMs use V_WMMA_F32_16X16X4_F32.
// All streaming traffic uses 128-bit accesses; attention stores are
// non-temporal so adj/h stay L2-resident.

typedef __attribute__((ext_vector_type(2))) float v2f;
typedef __attribute__((ext_vector_type(4))) float v4f;
typedef __attribute__((ext_vector_type(8))) float v8f;
typedef __attribute__((ext_vector_type(4))) int   v4i;

#define GAT_B   8
#define GAT_N   2048
#define GAT_IN  128
#define GAT_OUT 64
#define NEG_BIG (-9000000000000000.0f)
#define ALPHA   0.2f

// D(16x16,f32) = A(16x4,f32) * B(4x16,f32) + C
// A layout: lane&15 = M, lane half selects K pair {0,1}/{2,3}; vgpr0=Klo, vgpr1=Klo+1
// B layout (mirror): lane&15 = N, lane half selects K pair; vgpr0=Klo, vgpr1=Klo+1
// C/D layout: lane&15 = N, vgpr r -> M = r + 8*(lane>>4)
__device__ __forceinline__ v8f wmma4(v2f a, v2f b, v8f c) {
  return __builtin_amdgcn_wmma_f32_16x16x4_f32(
      /*neg_a=*/false, a, /*neg_b=*/false, b,
      /*c_mod=*/(short)0, c, /*reuse_a=*/false, /*reuse_b=*/false);
}

__device__ __forceinline__ float lrelu(float e) {
  return e > 0.0f ? e : ALPHA * e;
}

// ---------------------------------------------------------------------------
// Kernel 1: h = x @ W   (16384x128)*(128x64) -> (16384x64), fp32 WMMA.
// One wave per 16-row tile; 4 N-tiles of 16 accumulated in 4 v8f regs.
// ---------------------------------------------------------------------------
__global__ __launch_bounds__(32) void gat_gemm_h(const float* __restrict__ x,
                                                 const float* __restrict__ W,
                                                 float* __restrict__ h) {
  const int r0    = blockIdx.x * 16;
  const int l     = threadIdx.x;
  const int m16   = l & 15;
  const int khalf = (l >> 4) * 2;

  const float* xrow = x + (size_t)(r0 + m16) * GAT_IN;
  v8f c0 = {}, c1 = {}, c2 = {}, c3 = {};

  for (int k0 = 0; k0 < GAT_IN; k0 += 4) {
    v2f a = *reinterpret_cast<const v2f*>(xrow + k0 + khalf);
    const float* w0 = W + (size_t)(k0 + khalf) * GAT_OUT + m16;
    const float* w1 = w0 + GAT_OUT;
    v2f b0; b0.x = w0[0];  b0.y = w1[0];
    v2f b1; b1.x = w0[16]; b1.y = w1[16];
    v2f b2; b2.x = w0[32]; b2.y = w1[32];
    v2f b3; b3.x = w0[48]; b3.y = w1[48];
    c0 = wmma4(a, b0, c0);
    c1 = wmma4(a, b1, c1);
    c2 = wmma4(a, b2, c2);
    c3 = wmma4(a, b3, c3);
  }

  const int col = l & 15;
  const int rowoff = (l >> 4) * 8;
  for (int r = 0; r < 8; ++r) {
    float* hp = h + (size_t)(r0 + rowoff + r) * GAT_OUT + col;
    hp[0]  = c0[r];
    hp[16] = c1[r];
    hp[32] = c2[r];
    hp[48] = c3[r];
  }
}

// ---------------------------------------------------------------------------
// Kernel 2: si = h.a1, sj = h.a2 per row. One wave per row.
// ---------------------------------------------------------------------------
__global__ __launch_bounds__(256) void gat_scores(const float* __restrict__ h,
                                                  const float* __restrict__ a,
                                                  float* __restrict__ si,
                                                  float* __restrict__ sj) {
  const int row = blockIdx.x * 8 + (threadIdx.x >> 5);
  const int l   = threadIdx.x & 31;
  const float* hr = h + (size_t)row * GAT_OUT;
  const float h0 = hr[l], h1 = hr[l + 32];
  float p1 = h0 * a[l]           + h1 * a[l + 32];
  float p2 = h0 * a[GAT_OUT + l] + h1 * a[GAT_OUT + l + 32];
  for (int off = 16; off > 0; off >>= 1) {
    p1 += __shfl_xor(p1, off, 32);
    p2 += __shfl_xor(p2, off, 32);
  }
  if (l == 0) { si[row] = p1; sj[row] = p2; }
}

// ---------------------------------------------------------------------------
// Kernel 3 (fused): masked leakyrelu softmax -> write attention -> h' = att@h
// Block = 128 threads (4 waves) owns 16 rows of one batch.
//  Phase A/B: per-row max then sum(exp), streaming adj with b128 loads
//             (adj rows stay L2-hot across the 3 passes).
//  Phase C:   K chunks of 64: produce att chunk (b128 NT global store + LDS),
//             then each wave runs 16x v_wmma_f32_16x16x4_f32 on its 16-col tile.
// ---------------------------------------------------------------------------
__global__ __launch_bounds__(128) void gat_attn(const int*   __restrict__ adj,
                                                const float* __restrict__ h,
                                                const float* __restrict__ si,
                                                const float* __restrict__ sj,
                                                float* __restrict__ att_out,
                                                float* __restrict__ hprime) {
  constexpr int TM = 16, KC = 64, LDK = KC + 4;   // stride 68: 16B aligned,
  const int tile = blockIdx.x;                    // 4*lane bank stride (no conflicts)
  const int b    = tile >> 7;                     // 128 row-tiles per batch
  const int i0   = (tile & 127) * TM;
  const int tid  = threadIdx.x;
  const int wave = tid >> 5;
  const int lane = tid & 31;

  __shared__ float s_m[TM];
  __shared__ float s_inv[TM];
  __shared__ float s_att[TM][LDK];

  const int*   adjb = adj + ((size_t)b * GAT_N + i0) * GAT_N;
  const float* sjb  = sj + (size_t)b * GAT_N;
  const float* sib  = si + (size_t)b * GAT_N;
  const float* hb   = h  + (size_t)b * GAT_N * GAT_OUT;

  // ---- Phase A/B: per-row softmax stats (wave w handles rows w, w+4, ...)
  for (int r = wave; r < TM; r += 4) {
    const int* arow = adjb + (size_t)r * GAT_N;
    const float sival = sib[i0 + r];
    float m = NEG_BIG;
    for (int j = lane * 4; j < GAT_N; j += 128) {
      v4i av = *reinterpret_cast<const v4i*>(arow + j);
      v4f sv = *reinterpret_cast<const v4f*>(sjb + j);
      if (av.x > 0) m = fmaxf(m, lrelu(sival + sv.x));
      if (av.y > 0) m = fmaxf(m, lrelu(sival + sv.y));
      if (av.z > 0) m = fmaxf(m, lrelu(sival + sv.z));
      if (av.w > 0) m = fmaxf(m, lrelu(sival + sv.w));
    }
    for (int off = 16; off > 0; off >>= 1) m = fmaxf(m, __shfl_xor(m, off, 32));
    float s = 0.0f;
    for (int j = lane * 4; j < GAT_N; j += 128) {
      v4i av = *reinterpret_cast<const v4i*>(arow + j);
      v4f sv = *reinterpret_cast<const v4f*>(sjb + j);
      // masked -> exp(NEG_BIG - m) = 0; fully-masked row -> exp(0) = 1 (uniform)
      s += __expf((av.x > 0 ? lrelu(sival + sv.x) : NEG_BIG) - m);
      s += __expf((av.y > 0 ? lrelu(sival + sv.y) : NEG_BIG) - m);
      s += __expf((av.z > 0 ? lrelu(sival + sv.z) : NEG_BIG) - m);
      s += __expf((av.w > 0 ? lrelu(sival + sv.w) : NEG_BIG) - m);
    }
    for (int off = 16; off > 0; off >>= 1) s += __shfl_xor(s, off, 32);
    if (lane == 0) { s_m[r] = m; s_inv[r] = 1.0f / s; }
  }
  __syncthreads();

  // Producer mapping: thread covers rows pr0 = tid>>4 and pr0+8,
  // 4 consecutive cols c0 = (tid&15)*4 -> b128 loads/stores.
  const int pr0 = tid >> 4;
  const int pr1 = pr0 + 8;
  const int c0  = (tid & 15) * 4;
  const float pm0 = s_m[pr0], pi0 = s_inv[pr0], ps0 = sib[i0 + pr0];
  const float pm1 = s_m[pr1], pi1 = s_inv[pr1], ps1 = sib[i0 + pr1];

  // Consumer-side WMMA setup: wave owns cols [16*wave, 16*wave+16)
  const int m16   = lane & 15;
  const int khalf = (lane >> 4) * 2;
  const int nbase = wave * 16;
  v8f acc = {};

  for (int j0 = 0; j0 < GAT_N; j0 += KC) {
    // produce att[16][64]: b128 NT global store + LDS stage
    const int j = j0 + c0;
    const v4i a0 = *reinterpret_cast<const v4i*>(adjb + (size_t)pr0 * GAT_N + j);
    const v4i a1 = *reinterpret_cast<const v4i*>(adjb + (size_t)pr1 * GAT_N + j);
    const v4f sv = *reinterpret_cast<const v4f*>(sjb + j);
    v4f o0, o1;
    o0.x = __expf(((a0.x > 0) ? lrelu(ps0 + sv.x) : NEG_BIG) - pm0) * pi0;
    o0.y = __expf(((a0.y > 0) ? lrelu(ps0 + sv.y) : NEG_BIG) - pm0) * pi0;
    o0.z = __expf(((a0.z > 0) ? lrelu(ps0 + sv.z) : NEG_BIG) - pm0) * pi0;
    o0.w = __expf(((a0.w > 0) ? lrelu(ps0 + sv.w) : NEG_BIG) - pm0) * pi0;
    o1.x = __expf(((a1.x > 0) ? lrelu(ps1 + sv.x) : NEG_BIG) - pm1) * pi1;
    o1.y = __expf(((a1.y > 0) ? lrelu(ps1 + sv.y) : NEG_BIG) - pm1) * pi1;
    o1.z = __expf(((a1.z > 0) ? lrelu(ps1 + sv.z) : NEG_BIG) - pm1) * pi1;
    o1.w = __expf(((a1.w > 0) ? lrelu(ps1 + sv.w) : NEG_BIG) - pm1) * pi1;
    *reinterpret_cast<v4f*>(&s_att[pr0][c0]) = o0;
    *reinterpret_cast<v4f*>(&s_att[pr1][c0]) = o1;
    __builtin_nontemporal_store(
        o0, reinterpret_cast<v4f*>(att_out + ((size_t)b * GAT_N + i0 + pr0) * GAT_N + j));
    __builtin_nontemporal_store(
        o1, reinterpret_cast<v4f*>(att_out + ((size_t)b * GAT_N + i0 + pr1) * GAT_N + j));
    __syncthreads();

    // consume: acc(16x16) += att(16x64) @ h[j0:j0+64, nbase:nbase+16)
    const float* hK = hb + (size_t)j0 * GAT_OUT + nbase + m16;
    for (int kk = 0; kk < KC; kk += 4) {
      v2f a = *reinterpret_cast<const v2f*>(&s_att[m16][kk + khalf]);
      v2f bv;
      bv.x = hK[(size_t)(kk + khalf) * GAT_OUT];
      bv.y = hK[(size_t)(kk + khalf + 1) * GAT_OUT];
      acc = wmma4(a, bv, acc);
    }
    __syncthreads();
  }

  // store h'
  const int col = nbase + m16;
  const int rowoff = (lane >> 4) * 8;
  for (int r = 0; r < 8; ++r) {
    hprime[((size_t)b * GAT_N + i0 + rowoff + r) * GAT_OUT + col] = acc[r];
  }
}

// ---------------------------------------------------------------------------
extern "C" void kernel_launch(void* const* d_in, const int* in_sizes, int n_in,
                              void* d_out, int out_size, void* d_ws, size_t ws_size,
                              hipStream_t stream) {
  const float* x   = (const float*)d_in[0];   // (8,2048,128)
  const int*   adj = (const int*)  d_in[1];   // (8,2048,2048)
  const float* W   = (const float*)d_in[2];   // (128,64)
  const float* a   = (const float*)d_in[3];   // (128,1)

  // workspace: h (16384*64), si (16384), sj (16384)  ~4.33 MB
  float* h  = (float*)d_ws;
  float* si = h + (size_t)GAT_B * GAT_N * GAT_OUT;
  float* sj = si + (size_t)GAT_B * GAT_N;

  // outputs: h_prime (8*2048*64) then attention (8*2048*2048)
  float* hprime = (float*)d_out;
  float* att    = hprime + (size_t)GAT_B * GAT_N * GAT_OUT;

  gat_gemm_h<<<(GAT_B * GAT_N) / 16, 32, 0, stream>>>(x, W, h);
  gat_scores<<<(GAT_B * GAT_N) / 8, 256, 0, stream>>>(h, a, si, sj);
  gat_attn<<<(GAT_B * GAT_N) / 16, 128, 0, stream>>>(adj, h, si, sj, att, hprime);
}